// CustomLoss_8435315769872
// MI455X (gfx1250) — compile-verified
//
#include <hip/hip_runtime.h>

// CDNA5 / gfx1250. Bandwidth-bound streaming reduction:
//   134 MB read @ 23.3 TB/s  ->  ~5.8 us floor; compute (1 v_cos + 1 fma per
//   element) is far below the VALU/trans roofline, so the kernel is a single
//   coalesced pass. Wave-level reduction uses V_WMMA_F32_16X16X4_F32 with an
//   all-ones B matrix (D[m][n] = rowsum(A)) -- layout-robust matrix reduce.

typedef __attribute__((ext_vector_type(2))) float v2f;
typedef __attribute__((ext_vector_type(4))) float v4f;   // native clang vector -> NT b128 loads
typedef __attribute__((ext_vector_type(8))) float v8f;

#define THREADS 256
#define NCOLS   4096

// v_cos_f32 computes cos(2*pi*x) directly (input in revolutions).
// Reference needs cos(2pi(x-0.5))^2 == cos(2pi x)^2, so the shift drops out.
__device__ __forceinline__ float cos2pi_sq(float x) {
  float c = __builtin_amdgcn_cosf(x);
  return c * c;
}

__device__ __forceinline__ float wave32_sum_wmma(float p) {
  // A (16x4 f32, 2 VGPRs/lane): slot0 = partial, slot1 = 0.
  // B = all-ones -> D[m][n] = sum over K of A[m][k].
  // Sum of all 16 D rows == sum of ALL A slots == sum of the 32 lane partials,
  // independent of the exact K-striping. Each lane holds 8 D rows; the other
  // 8 rows live in lane^16, so one xor-16 shuffle completes the reduction.
  v2f a; a[0] = p;    a[1] = 0.0f;
  v2f b; b[0] = 1.0f; b[1] = 1.0f;
  v8f c = {};
  v8f d = __builtin_amdgcn_wmma_f32_16x16x4_f32(
      /*neg_a=*/false, a, /*neg_b=*/false, b,
      /*c_mod=*/(short)0, c, /*reuse_a=*/false, /*reuse_b=*/false);
  float r = d[0] + d[1] + d[2] + d[3] + d[4] + d[5] + d[6] + d[7];
  r += __shfl_xor(r, 16, 32);
  return r;
}

__global__ __launch_bounds__(THREADS) void rowprod_kernel(
    const float* __restrict__ o, const float* __restrict__ t,
    float* __restrict__ rowprod) {
  const int row = blockIdx.x;
  const int tid = threadIdx.x;
  const size_t base = (size_t)row * NCOLS;
  const v4f* o4 = reinterpret_cast<const v4f*>(o + base);
  const v4f* t4 = reinterpret_cast<const v4f*>(t + base);

  float so = 0.0f, st = 0.0f;
#pragma unroll
  for (int k = 0; k < NCOLS / (4 * THREADS); ++k) {
    // Lane-contiguous 128-bit loads: each wave covers 512B per instruction.
    v4f a = __builtin_nontemporal_load(&o4[tid + k * THREADS]);
    v4f b = __builtin_nontemporal_load(&t4[tid + k * THREADS]);
    so += cos2pi_sq(a[0]); so += cos2pi_sq(a[1]);
    so += cos2pi_sq(a[2]); so += cos2pi_sq(a[3]);
    st += cos2pi_sq(b[0]); st += cos2pi_sq(b[1]);
    st += cos2pi_sq(b[2]); st += cos2pi_sq(b[3]);
  }

  // Per-wave32 totals via WMMA reduction (EXEC is all-1s here: no divergence).
  float ro = wave32_sum_wmma(so);
  float rt = wave32_sum_wmma(st);

  __shared__ float lso[THREADS / 32];
  __shared__ float lst[THREADS / 32];
  const int lane = tid & 31;
  const int wv   = tid >> 5;
  if (lane == 0) { lso[wv] = ro; lst[wv] = rt; }
  __syncthreads();
  if (tid == 0) {
    float SO = 0.0f, ST = 0.0f;
#pragma unroll
    for (int w = 0; w < THREADS / 32; ++w) { SO += lso[w]; ST += lst[w]; }
    rowprod[row] = __builtin_sqrtf(SO) * __builtin_sqrtf(ST);
  }
}

__global__ __launch_bounds__(THREADS) void final_reduce_kernel(
    const float* __restrict__ rowprod, float* __restrict__ out, int n) {
  float s = 0.0f;
  for (int i = threadIdx.x; i < n; i += THREADS) s += rowprod[i];
  // Deterministic tree reduction (no float atomics -> bitwise-stable replays).
  s = wave32_sum_wmma(s);
  __shared__ float lds[THREADS / 32];
  const int lane = threadIdx.x & 31;
  const int wv   = threadIdx.x >> 5;
  if (lane == 0) lds[wv] = s;
  __syncthreads();
  if (threadIdx.x == 0) {
    float tot = 0.0f;
#pragma unroll
    for (int w = 0; w < THREADS / 32; ++w) tot += lds[w];
    out[0] = tot;
  }
}

extern "C" void kernel_launch(void* const* d_in, const int* in_sizes, int n_in,
                              void* d_out, int out_size, void* d_ws, size_t ws_size,
                              hipStream_t stream) {
  const float* o = (const float*)d_in[0];   // "output" [B, N] fp32
  const float* t = (const float*)d_in[1];   // "target" [B, N] fp32
  float* rowprod = (float*)d_ws;            // B floats of scratch
  float* out     = (float*)d_out;           // scalar fp32

  const int nrows = in_sizes[0] / NCOLS;    // B = 4096

  rowprod_kernel<<<nrows, THREADS, 0, stream>>>(o, t, rowprod);
  final_reduce_kernel<<<1, THREADS, 0, stream>>>(rowprod, out, nrows);
}